// MeanAggregator_6854767804435
// MI455X (gfx1250) — compile-verified
//
#include <hip/hip_runtime.h>
#include <hip/hip_bf16.h>
#include <stdint.h>

typedef __attribute__((ext_vector_type(2))) float v2f;
typedef __attribute__((ext_vector_type(8))) float v8f;

#define D_FEAT 128
#define LDS_STRIDE 132   // 128 + 4 dwords pad -> conflict-free column reads (132 % 64 = 4)

// ---------------------------------------------------------------------------
// Kernel 1: self-edge init.  out[i][:] = x[i][:],  counts[i] = 1
// ---------------------------------------------------------------------------
__global__ void mean_agg_init(const float* __restrict__ x,
                              float* __restrict__ outsum,
                              float* __restrict__ counts,
                              int n_nodes) {
    long long idx = (long long)blockIdx.x * blockDim.x + threadIdx.x;
    long long total = (long long)n_nodes * D_FEAT;
    if (idx < total) outsum[idx] = x[idx];
    if (idx < n_nodes) counts[idx] = 1.0f;
}

// ---------------------------------------------------------------------------
// Kernel 2: edge scatter-add.  One wave32 per edge: lane l handles 4 floats
// (float4 gather, coalesced 512B per wave; 4x global_atomic_add_f32 scatter).
// input_matrix (51.2 MB) and the accumulator (51.2 MB) both fit in the
// 192 MB L2, so this random-access phase is L2-resident.  All offsets are
// 32-bit (N*128 = 12.8M < 2^31) so addressing is saddr + 32-bit voffset.
// ---------------------------------------------------------------------------
__global__ void mean_agg_scatter(const float* __restrict__ x,
                                 const long long* __restrict__ adj,  // [2, E] row-major
                                 float* __restrict__ outsum,
                                 float* __restrict__ counts,
                                 int n_edges) {
    int wave = (int)(((long long)blockIdx.x * blockDim.x + threadIdx.x) >> 5);
    int lane = threadIdx.x & 31;
    if (wave >= n_edges) return;

    uint32_t dst = (uint32_t)adj[wave];                       // row 0: segment ids
    uint32_t nbr = (uint32_t)adj[(long long)n_edges + wave];  // row 1: gather indices

    const float4 v = *(const float4*)(x + nbr * (uint32_t)D_FEAT + (uint32_t)(lane * 4));

    float* o = outsum + dst * (uint32_t)D_FEAT + (uint32_t)(lane * 4);
    atomicAdd(o + 0, v.x);                           // -> global_atomic_add_f32
    atomicAdd(o + 1, v.y);
    atomicAdd(o + 2, v.z);
    atomicAdd(o + 3, v.w);
    if (lane == 0) atomicAdd(&counts[dst], 1.0f);
}

// ---------------------------------------------------------------------------
// Kernel 3: divide by counts + (mean @ W) with V_WMMA_F32_16X16X4_F32.
// Block = 256 threads = 8 waves; block owns 16 output rows, each wave owns
// one 16-column tile of W; K=128 -> 32 WMMA ops per wave, full f32 accuracy.
// In-place over d_out: rows are staged into LDS behind a barrier first.
// Full-tile fast path (no per-element guards); NT load on the last-use sums,
// NT store on the write-once result to keep L2 for the hot 51 MB working set.
// ---------------------------------------------------------------------------
__global__ void mean_agg_gemm(const float* __restrict__ Wm,     // [128,128]
                              const float* __restrict__ counts,
                              float* __restrict__ out,          // in: sums, out: result
                              int n_nodes) {
    __shared__ float smem[16 * LDS_STRIDE];
    __shared__ float inv[16];

    const int r0   = blockIdx.x * 16;
    const int tid  = threadIdx.x;
    const bool full = (r0 + 16 <= n_nodes);

    if (tid < 16) {
        int r = r0 + tid;
        inv[tid] = (r < n_nodes) ? (1.0f / counts[r]) : 0.0f;
    }
    __syncthreads();

    // Stage 16x128 rows of the sum into LDS, scaled to the mean (last-use -> NT).
    const float* __restrict__ srcp = out + (uint32_t)r0 * (uint32_t)D_FEAT;
    if (full) {
        for (int i = tid; i < 16 * D_FEAT; i += 256) {
            int row = i >> 7;
            int col = i & 127;
            float v = __builtin_nontemporal_load(srcp + i);
            smem[row * LDS_STRIDE + col] = v * inv[row];
        }
    } else {
        for (int i = tid; i < 16 * D_FEAT; i += 256) {
            int row = i >> 7;
            int col = i & 127;
            float v = (r0 + row < n_nodes) ? __builtin_nontemporal_load(srcp + i) : 0.0f;
            smem[row * LDS_STRIDE + col] = v * inv[row];
        }
    }
    __syncthreads();

    const int wv   = tid >> 5;          // wave id 0..7  -> N tile
    const int lane = tid & 31;
    const int n0   = wv * 16;
    const int m    = lane & 15;
    const int klo  = (lane >> 4) << 1;  // lanes 0-15: K=0,1 ; lanes 16-31: K=2,3

    v8f c = {};
    #pragma unroll 4
    for (int kb = 0; kb < D_FEAT; kb += 4) {
        v2f a, b;
        // A fragment (16x4 f32, ISA layout): lane m holds A[m][kb+klo], A[m][kb+klo+1]
        a.x = smem[m * LDS_STRIDE + kb + klo];
        a.y = smem[m * LDS_STRIDE + kb + klo + 1];
        // B fragment (4x16 f32): lane holds W[kb+klo][n0+m], W[kb+klo+1][n0+m]
        b.x = Wm[(kb + klo)     * D_FEAT + n0 + m];
        b.y = Wm[(kb + klo + 1) * D_FEAT + n0 + m];
        c = __builtin_amdgcn_wmma_f32_16x16x4_f32(
                /*neg_a=*/false, a, /*neg_b=*/false, b,
                /*c_mod=*/(short)0, c, /*reuse_a=*/false, /*reuse_b=*/false);
    }

    // D layout: VGPR i -> M = i (lanes 0-15) or i+8 (lanes 16-31), N = lane&15
    const int mb = r0 + ((lane >> 4) << 3);
    float* __restrict__ dstp =
        out + (uint32_t)mb * (uint32_t)D_FEAT + (uint32_t)(n0 + m);
    if (full) {
        #pragma unroll
        for (int i = 0; i < 8; ++i)
            __builtin_nontemporal_store(c[i], dstp + i * D_FEAT);
    } else {
        #pragma unroll
        for (int i = 0; i < 8; ++i)
            if (mb + i < n_nodes)
                __builtin_nontemporal_store(c[i], dstp + i * D_FEAT);
    }
}

// ---------------------------------------------------------------------------
extern "C" void kernel_launch(void* const* d_in, const int* in_sizes, int n_in,
                              void* d_out, int out_size, void* d_ws, size_t ws_size,
                              hipStream_t stream) {
    (void)n_in; (void)out_size; (void)ws_size;

    const float*     x   = (const float*)d_in[0];       // [N,128] f32
    const long long* adj = (const long long*)d_in[1];   // [2,E] int64
    const float*     Wm  = (const float*)d_in[2];       // [128,128] f32
    float*           out = (float*)d_out;               // [N,128] f32
    float*           cnt = (float*)d_ws;                // [N] f32 scratch

    const int n_nodes = in_sizes[0] / D_FEAT;
    const int n_edges = in_sizes[1] / 2;

    // Phase 1: self-edge init
    {
        long long total = (long long)n_nodes * D_FEAT;
        int blocks = (int)((total + 255) / 256);
        mean_agg_init<<<blocks, 256, 0, stream>>>(x, out, cnt, n_nodes);
    }
    // Phase 2: edge scatter (8 edges per 256-thread block, one wave32 per edge)
    {
        int blocks = (n_edges + 7) / 8;
        mean_agg_scatter<<<blocks, 256, 0, stream>>>(x, adj, out, cnt, n_edges);
    }
    // Phase 3: mean + WMMA projection, 16 rows per block
    {
        int blocks = (n_nodes + 15) / 16;
        mean_agg_gemm<<<blocks, 256, 0, stream>>>(Wm, cnt, out, n_nodes);
    }
}